// CustomBRepEncoder_39840116638566
// MI455X (gfx1250) — compile-verified
//
#include <hip/hip_runtime.h>
#include <float.h>

#define NEG_SLOPE 0.01f

typedef __attribute__((ext_vector_type(2))) float v2f;
typedef __attribute__((ext_vector_type(4))) float v4f;
typedef __attribute__((ext_vector_type(8))) float v8f;

__device__ __forceinline__ float leaky(float x) {
    return x > 0.0f ? x : NEG_SLOPE * x;
}

// ---------------------------------------------------------------------------
// out[N,64] = leaky(X[N,K] @ W[K,64] + b), K in {3,15,17}
// block = 256 threads -> 4 rows x 64 cols; weights staged in LDS.
// ---------------------------------------------------------------------------
__global__ __launch_bounds__(256) void embed_kernel(
    const float* __restrict__ X, const float* __restrict__ Wm,
    const float* __restrict__ b, float* __restrict__ out, int K)
{
    __shared__ float Wl[17 * 64];
    __shared__ float bl[64];
    int tid = threadIdx.x;
    for (int i = tid; i < K * 64; i += 256) Wl[i] = Wm[i];
    if (tid < 64) bl[tid] = b[tid];
    __syncthreads();

    long row = (long)blockIdx.x * 4 + (tid >> 6);
    int  col = tid & 63;
    float acc = bl[col];
    const float* xr = X + row * K;
#pragma unroll 4
    for (int k = 0; k < K; ++k) acc += xr[k] * Wl[k * 64 + col];
    out[row * 64 + col] = leaky(acc);
}

// ---------------------------------------------------------------------------
// fill n4 float4's with value v (scatter-max init to -FLT_MAX)
// ---------------------------------------------------------------------------
__global__ __launch_bounds__(256) void fill_kernel(float* __restrict__ p, float v, long n4)
{
    long i = (long)blockIdx.x * blockDim.x + threadIdx.x;
    if (i < n4) {
        v4f val = {v, v, v, v};
        reinterpret_cast<v4f*>(p)[i] = val;
    }
}

// ---------------------------------------------------------------------------
// For each incidence e: maxes[edst[e]] = max(maxes[edst[e]],
//                                            xdst[edst[e]] - xsrc[esrc[e]])
// 16 threads per edge, float4 per thread -> global_atomic_max_num_f32 x4.
// ---------------------------------------------------------------------------
__global__ __launch_bounds__(256) void scatter_kernel(
    const float* __restrict__ xsrc, const float* __restrict__ xdst,
    const int* __restrict__ esrc, const int* __restrict__ edst,
    float* __restrict__ maxes, int E)
{
    long t = (long)blockIdx.x * blockDim.x + threadIdx.x;
    int e = (int)(t >> 4);
    if (e >= E) return;
    int c = (int)(t & 15) * 4;
    long d = (long)edst[e] * 64 + c;
    long s = (long)esrc[e] * 64 + c;
    v4f a  = *reinterpret_cast<const v4f*>(xdst + d);
    v4f bb = *reinterpret_cast<const v4f*>(xsrc + s);
    atomicMax(&maxes[d + 0], a.x - bb.x);
    atomicMax(&maxes[d + 1], a.y - bb.y);
    atomicMax(&maxes[d + 2], a.z - bb.z);
    atomicMax(&maxes[d + 3], a.w - bb.w);
}

// ---------------------------------------------------------------------------
// out[N,64] = xdst + leaky(concat(xdst, clean(maxes)) @ Wm[128,64] + bias)
// clean(): -FLT_MAX sentinel (empty segment) -> 0.
//
// Block = 256 threads = 8 waves; each wave computes 16 rows x 64 cols via
// V_WMMA_F32_16X16X4_F32 (K=128 over 32 k-steps, 4 N-tiles).
//
// LDS holds the weight matrix pre-swizzled into per-lane B fragments:
//   Wl[kk][nt][lane][2] with
//     frag = { Wm[kk*4 + 2*(lane>>4) + 0][nt*16 + (lane&15)],
//              Wm[kk*4 + 2*(lane>>4) + 1][nt*16 + (lane&15)] }
// so each B operand is one contiguous ds_load_b64 into an aligned VGPR pair.
//
// A fragment per lane (16x4 f32 layout): row = rowbase + (lane&15),
//   {K = k0 + 2*half, K = k0 + 2*half + 1}  -> one global float2 load.
// C/D tile: VGPR i holds (M=i, N=lane&15) for lanes 0-15, (M=i+8) for 16-31.
// ---------------------------------------------------------------------------
__global__ __launch_bounds__(256) void conv_kernel(
    const float* __restrict__ xdst, const float* __restrict__ maxes,
    const float* __restrict__ Wm /*128x64*/, const float* __restrict__ bias /*64*/,
    float* __restrict__ out)
{
    __shared__ float Wl[32 * 4 * 32 * 2];   // 8192 floats = 32 KB
    __shared__ float bl[64];
    int tid = threadIdx.x;

    // build swizzled B fragments (4096 fragments, 2 floats each)
    for (int idx = tid; idx < 4096; idx += 256) {
        int lane_ = idx & 31;
        int nt_   = (idx >> 5) & 3;
        int kk_   = idx >> 7;
        int ka_   = kk_ * 4 + (lane_ >> 4) * 2;
        int col_  = nt_ * 16 + (lane_ & 15);
        Wl[idx * 2 + 0] = Wm[ka_ * 64 + col_];
        Wl[idx * 2 + 1] = Wm[(ka_ + 1) * 64 + col_];
    }
    if (tid < 64) bl[tid] = bias[tid];
    __syncthreads();

    int wave = tid >> 5;
    int lane = tid & 31;
    int half = lane >> 4;
    int r    = lane & 15;

    long rowbase = (long)blockIdx.x * 128 + wave * 16;
    const float* xrow = xdst  + (rowbase + r) * 64 + half * 2;
    const float* mrow = maxes + (rowbase + r) * 64 + half * 2;
    const v2f*   frag = reinterpret_cast<const v2f*>(Wl) + lane;  // + kk*128 + nt*32

    v8f c0 = {}, c1 = {}, c2 = {}, c3 = {};

    // K = 0..63 : A from xdst (first half of the concat)
#pragma unroll
    for (int kk = 0; kk < 16; ++kk) {
        v2f a  = *reinterpret_cast<const v2f*>(xrow + kk * 4);
        v2f b0 = frag[kk * 128 +  0];
        v2f b1 = frag[kk * 128 + 32];
        v2f b2 = frag[kk * 128 + 64];
        v2f b3 = frag[kk * 128 + 96];
        c0 = __builtin_amdgcn_wmma_f32_16x16x4_f32(false, a, false, b0, (short)0, c0, false, false);
        c1 = __builtin_amdgcn_wmma_f32_16x16x4_f32(false, a, false, b1, (short)0, c1, false, false);
        c2 = __builtin_amdgcn_wmma_f32_16x16x4_f32(false, a, false, b2, (short)0, c2, false, false);
        c3 = __builtin_amdgcn_wmma_f32_16x16x4_f32(false, a, false, b3, (short)0, c3, false, false);
    }

    // K = 64..127 : A from maxes (second half), single-use -> non-temporal
#pragma unroll
    for (int kk = 16; kk < 32; ++kk) {
        v2f a = __builtin_nontemporal_load(
                    reinterpret_cast<const v2f*>(mrow + (kk - 16) * 4));
        a.x = (a.x == -FLT_MAX) ? 0.0f : a.x;   // empty segment -> 0
        a.y = (a.y == -FLT_MAX) ? 0.0f : a.y;
        v2f b0 = frag[kk * 128 +  0];
        v2f b1 = frag[kk * 128 + 32];
        v2f b2 = frag[kk * 128 + 64];
        v2f b3 = frag[kk * 128 + 96];
        c0 = __builtin_amdgcn_wmma_f32_16x16x4_f32(false, a, false, b0, (short)0, c0, false, false);
        c1 = __builtin_amdgcn_wmma_f32_16x16x4_f32(false, a, false, b1, (short)0, c1, false, false);
        c2 = __builtin_amdgcn_wmma_f32_16x16x4_f32(false, a, false, b2, (short)0, c2, false, false);
        c3 = __builtin_amdgcn_wmma_f32_16x16x4_f32(false, a, false, b3, (short)0, c3, false, false);
    }

    // epilogue: residual + bias + leaky
    v8f cs[4] = {c0, c1, c2, c3};
#pragma unroll
    for (int nt = 0; nt < 4; ++nt) {
        int col = nt * 16 + r;
        float bv = bl[col];
#pragma unroll
        for (int i = 0; i < 8; ++i) {
            long row = rowbase + i + half * 8;
            long idx = row * 64 + col;
            out[idx] = xdst[idx] + leaky(cs[nt][i] + bv);
        }
    }
}

// ---------------------------------------------------------------------------
extern "C" void kernel_launch(void* const* d_in, const int* in_sizes, int n_in,
                              void* d_out, int out_size, void* d_ws, size_t ws_size,
                              hipStream_t stream)
{
    (void)n_in; (void)out_size; (void)ws_size;

    const float* vertices = (const float*)d_in[0];
    const float* edges    = (const float*)d_in[1];
    const float* faces    = (const float*)d_in[2];
    const float* Wv       = (const float*)d_in[3];
    const float* bv       = (const float*)d_in[4];
    const float* We       = (const float*)d_in[5];
    const float* be       = (const float*)d_in[6];
    const float* Wf       = (const float*)d_in[7];
    const float* bf       = (const float*)d_in[8];
    const float* W_v2e    = (const float*)d_in[9];
    const float* b_v2e    = (const float*)d_in[10];
    const float* W_e2f    = (const float*)d_in[11];
    const float* b_e2f    = (const float*)d_in[12];
    const float* W_msg    = (const float*)d_in[13];
    const float* b_msg    = (const float*)d_in[14];
    const int*   e2v      = (const int*)d_in[15];
    const int*   f2e      = (const int*)d_in[16];
    const int*   f2f      = (const int*)d_in[17];

    const int N      = in_sizes[0] / 3;     // 262144
    const int E_e2v  = in_sizes[15] / 2;    // 524288
    const int E_f2e  = in_sizes[16] / 2;    // 1048576
    const int E_f2f  = in_sizes[17] / 2;    // 1048576

    float* x_v = (float*)d_ws;
    float* x_e = x_v + (size_t)N * 64;
    float* x_f = x_e + (size_t)N * 64;
    float* mx  = x_f + (size_t)N * 64;

    dim3 blk(256);
    long n4 = (long)N * 64 / 4;
    int fillGrid = (int)((n4 + 255) / 256);

    // input embeddings
    embed_kernel<<<N / 4, blk, 0, stream>>>(vertices, Wv, bv, x_v, 3);
    embed_kernel<<<N / 4, blk, 0, stream>>>(edges,    We, be, x_e, 15);
    embed_kernel<<<N / 4, blk, 0, stream>>>(faces,    Wf, bf, x_f, 17);

    // V2E: src = x_v (e2v row1), dst = x_e (e2v row0)
    fill_kernel<<<fillGrid, blk, 0, stream>>>(mx, -FLT_MAX, n4);
    scatter_kernel<<<(E_e2v * 16 + 255) / 256, blk, 0, stream>>>(x_v, x_e, e2v + E_e2v, e2v, mx, E_e2v);
    conv_kernel<<<N / 128, blk, 0, stream>>>(x_e, mx, W_v2e, b_v2e, x_e);

    // E2F: src = x_e (f2e row1), dst = x_f (f2e row0)
    fill_kernel<<<fillGrid, blk, 0, stream>>>(mx, -FLT_MAX, n4);
    scatter_kernel<<<(E_f2e * 16 + 255) / 256, blk, 0, stream>>>(x_e, x_f, f2e + E_f2e, f2e, mx, E_f2e);
    conv_kernel<<<N / 128, blk, 0, stream>>>(x_f, mx, W_e2f, b_e2f, x_f);

    // F2F layer 0: src idx = f2f row0, dst idx = f2f row1
    fill_kernel<<<fillGrid, blk, 0, stream>>>(mx, -FLT_MAX, n4);
    scatter_kernel<<<(E_f2f * 16 + 255) / 256, blk, 0, stream>>>(x_f, x_f, f2f, f2f + E_f2f, mx, E_f2f);
    conv_kernel<<<N / 128, blk, 0, stream>>>(x_f, mx, W_msg, b_msg, x_f);

    // F2F layer 1 -> final output
    fill_kernel<<<fillGrid, blk, 0, stream>>>(mx, -FLT_MAX, n4);
    scatter_kernel<<<(E_f2f * 16 + 255) / 256, blk, 0, stream>>>(x_f, x_f, f2f, f2f + E_f2f, mx, E_f2f);
    conv_kernel<<<N / 128, blk, 0, stream>>>(x_f, mx, W_msg + 128 * 64, b_msg + 64, (float*)d_out);
}